// DGCNNVanillaGroupNorm_21406117004178
// MI455X (gfx1250) — compile-verified
//
#include <hip/hip_runtime.h>
#include <hip/hip_bf16.h>
#include <float.h>

typedef __attribute__((ext_vector_type(2))) float v2f;
typedef __attribute__((ext_vector_type(8))) float v8f;

#define KNN 20
#define EPSV 1e-5f

// ---------- order-preserving float<->uint encoding for atomic max/min ----------
__device__ __forceinline__ unsigned fenc(float f) {
  unsigned u = __float_as_uint(f);
  return (u & 0x80000000u) ? ~u : (u | 0x80000000u);
}
__device__ __forceinline__ float fdec(unsigned u) {
  return (u & 0x80000000u) ? __uint_as_float(u & 0x7fffffffu) : __uint_as_float(~u);
}
__device__ __forceinline__ float leakyf(float v) { return v >= 0.f ? v : 0.2f * v; }

__device__ __forceinline__ v2f ld2(const float* p) {
  float2 t = *(const float2*)p;
  v2f r; r[0] = t.x; r[1] = t.y; return r;
}

// wave-level sum reduce (lanes 0..15 / 16..31 independently) + atomic per half-leader
__device__ __forceinline__ void stats_reduce_atomic(float s, float ss, int l, int h,
                                                    int o0, int CG, int b, int G,
                                                    float* __restrict__ stats) {
#pragma unroll
  for (int m = 1; m < 16; m <<= 1) {
    s  += __shfl_xor(s,  m, 32);
    ss += __shfl_xor(ss, m, 32);
  }
  if (l == 0) {
    int g = (o0 + 8 * h) / CG;   // all 8 rows of a half-lane share one group
    atomicAdd(&stats[2 * ((size_t)b * G + g) + 0], s);
    atomicAdd(&stats[2 * ((size_t)b * G + g) + 1], ss);
  }
}

// ============================================================================
// util kernels
// ============================================================================
__global__ void k_zero_f(float* p, int n) {
  int i = blockIdx.x * blockDim.x + threadIdx.x;
  if (i < n) p[i] = 0.f;
}

__global__ void k_init_mm(unsigned* mm, int half, int n) {
  int i = blockIdx.x * blockDim.x + threadIdx.x;
  if (i < n) { mm[i] = 0u; mm[half + i] = 0xFFFFFFFFu; }
}

// x (B,3,N) -> xt (B,N,3)
__global__ void k_transpose3(const float* __restrict__ x, float* __restrict__ xt,
                             int B, int N) {
  int i = blockIdx.x * blockDim.x + threadIdx.x;
  int total = B * 3 * N;
  if (i >= total) return;
  int n = i % N; int c = (i / N) % 3; int b = i / (3 * N);
  xt[((size_t)b * N + n) * 3 + c] = x[i];
}

// ============================================================================
// kNN: one wave per query point.  Per-lane sorted top-K insert, LDS merge.
// xt is point-major (B,N,C).
// ============================================================================
template<int C>
__global__ void k_knn(const float* __restrict__ xt, int* __restrict__ idx,
                      int B, int N) {
  __shared__ float ctr[8][C];
  __shared__ float mdist[8][32 * KNN];
  __shared__ int   midx [8][32 * KNN];
  int w    = threadIdx.x >> 5;
  int lane = threadIdx.x & 31;
  int p = blockIdx.x * 8 + w;
  int b = p / N, n = p % N;
  const float* base = xt + (size_t)b * N * C;
  for (int c = lane; c < C; c += 32) ctr[w][c] = base[(size_t)n * C + c];
  __syncthreads();

  float bd[KNN]; int bi[KNN];
#pragma unroll
  for (int q = 0; q < KNN; q++) { bd[q] = FLT_MAX; bi[q] = 0; }

  for (int m = lane; m < N; m += 32) {
    const float* pm = base + (size_t)m * C;
    __builtin_prefetch(pm + 32 * C, 0, 1);   // global_prefetch_b8
    float dist = 0.f;
    if (C % 4 == 0) {
#pragma unroll
      for (int c = 0; c < C; c += 4) {
        float4 v = *(const float4*)(pm + c);
        float d0 = v.x - ctr[w][c + 0], d1 = v.y - ctr[w][c + 1];
        float d2 = v.z - ctr[w][c + 2], d3 = v.w - ctr[w][c + 3];
        dist += d0 * d0 + d1 * d1 + d2 * d2 + d3 * d3;
      }
    } else {
#pragma unroll
      for (int c = 0; c < C; c++) { float d = pm[c] - ctr[w][c]; dist += d * d; }
    }
    if (dist < bd[KNN - 1]) {
      float dv = dist; int iv = m;
#pragma unroll
      for (int q = 0; q < KNN; q++) {
        if (dv < bd[q]) {
          float tf = bd[q]; bd[q] = dv; dv = tf;
          int   ti = bi[q]; bi[q] = iv; iv = ti;
        }
      }
    }
  }
#pragma unroll
  for (int q = 0; q < KNN; q++) {
    mdist[w][lane * KNN + q] = bd[q];
    midx [w][lane * KNN + q] = bi[q];
  }
  __syncthreads();
  if (lane == 0) {
    int head[32];
    for (int t = 0; t < 32; t++) head[t] = 0;
    int* op = idx + ((size_t)b * N + n) * KNN;
    for (int j = 0; j < KNN; j++) {
      float best = FLT_MAX; int bt = 0;
      for (int t = 0; t < 32; t++) {
        float dv = (head[t] < KNN) ? mdist[w][t * KNN + head[t]] : FLT_MAX;
        if (dv < best) { best = dv; bt = t; }
      }
      op[j] = midx[w][bt * KNN + head[bt]];
      head[bt]++;
    }
  }
  __syncthreads();
}

// ============================================================================
// Edge-conv GEMM:  h(O x N*K) = W(O x 2C) @ f ,  f col (n,k):
//   c <  C : xt[b, idx[n,k], c] - xt[b, n, c]
//   c >= C : xt[b, n, c-C]
// One wave -> 16(O) x 64(cols): 4 accumulators, A fragment reused 4x.
// K-loop split into two uniform halves (diff part / center part) => no
// lane-divergent guards, straight float2 fragment loads.
// Accumulates group stats + per-(o,n) running max/min over k (encoded uints).
// ============================================================================
template<int C>
__global__ void k_edge_gemm(const float* __restrict__ W,
                            const float* __restrict__ xt,
                            const int* __restrict__ idx,
                            int O, int N,
                            float* __restrict__ stats, int CG,
                            unsigned* __restrict__ mmax, unsigned* __restrict__ mmin) {
  constexpr int C2 = 2 * C;
  int b  = blockIdx.z;
  int o0 = blockIdx.y * 16;
  int w    = threadIdx.x >> 5;
  int lane = threadIdx.x & 31;
  int h = lane >> 4, l = lane & 15;
  int col0 = (blockIdx.x * 8 + w) * 64;         // N*K divisible by 64*8

  int nT[4];
  const float* pjT[4];
  const float* pcT[4];
#pragma unroll
  for (int t = 0; t < 4; t++) {
    int col = col0 + 16 * t + l;
    int n = col / KNN;
    int k = col - n * KNN;
    int j = idx[((size_t)b * N + n) * KNN + k];
    nT[t]  = n;
    pcT[t] = xt + ((size_t)b * N + n) * C;
    pjT[t] = xt + ((size_t)b * N + j) * C;
  }
  const float* wrow = W + (size_t)(o0 + l) * C2;

  v8f acc[4];
#pragma unroll
  for (int t = 0; t < 4; t++) acc[t] = (v8f){};

  if constexpr (C == 3) {
    // 2C = 6: fully unrolled, branchless operand selection on lane-half h.
    float w0 = wrow[0], w1 = wrow[1], w2 = wrow[2];
    float w3 = wrow[3], w4 = wrow[4], w5 = wrow[5];
    v2f a0, a1;
    a0[0] = h ? w2 : w0;  a0[1] = h ? w3 : w1;
    a1[0] = h ? 0.f : w4; a1[1] = h ? 0.f : w5;
#pragma unroll
    for (int t = 0; t < 4; t++) {
      float f0 = pjT[t][0] - pcT[t][0];
      float f1 = pjT[t][1] - pcT[t][1];
      float f2 = pjT[t][2] - pcT[t][2];
      float f3 = pcT[t][0], f4 = pcT[t][1], f5 = pcT[t][2];
      v2f b0, b1;
      b0[0] = h ? f2  : f0;  b0[1] = h ? f3  : f1;
      b1[0] = h ? 0.f : f4;  b1[1] = h ? 0.f : f5;
      acc[t] = __builtin_amdgcn_wmma_f32_16x16x4_f32(false, a0, false, b0,
                                                     (short)0, acc[t], false, false);
      acc[t] = __builtin_amdgcn_wmma_f32_16x16x4_f32(false, a1, false, b1,
                                                     (short)0, acc[t], false, false);
    }
  } else {
    // first half: channels [0,C)  ->  pj - pc   (ka = k0+2h <= C-2 always)
    for (int k0 = 0; k0 < C; k0 += 4) {
      int ka = k0 + 2 * h;
      v2f a = ld2(wrow + ka);
#pragma unroll
      for (int t = 0; t < 4; t++) {
        v2f pj2 = ld2(pjT[t] + ka);
        v2f pc2 = ld2(pcT[t] + ka);
        v2f bb; bb[0] = pj2[0] - pc2[0]; bb[1] = pj2[1] - pc2[1];
        acc[t] = __builtin_amdgcn_wmma_f32_16x16x4_f32(false, a, false, bb,
                                                       (short)0, acc[t], false, false);
      }
    }
    // second half: channels [C,2C) ->  pc[c-C]
    for (int k0 = 0; k0 < C; k0 += 4) {
      int ka = k0 + 2 * h;
      v2f a = ld2(wrow + C + ka);
#pragma unroll
      for (int t = 0; t < 4; t++) {
        v2f bb = ld2(pcT[t] + ka);
        acc[t] = __builtin_amdgcn_wmma_f32_16x16x4_f32(false, a, false, bb,
                                                       (short)0, acc[t], false, false);
      }
    }
  }

  // ---- epilogue: group stats + per-(o,n) max/min ----
  float s = 0.f, ss = 0.f;
#pragma unroll
  for (int t = 0; t < 4; t++)
#pragma unroll
    for (int v = 0; v < 8; v++) { float d = acc[t][v]; s += d; ss += d * d; }
  stats_reduce_atomic(s, ss, l, h, o0, CG, b, O / CG, stats);

#pragma unroll
  for (int t = 0; t < 4; t++) {
    size_t rowBase = ((size_t)b * O + (o0 + 8 * h)) * N + nT[t];
#pragma unroll
    for (int v = 0; v < 8; v++) {
      unsigned e = fenc(acc[t][v]);
      atomicMax(&mmax[rowBase + (size_t)v * N], e);
      atomicMin(&mmin[rowBase + (size_t)v * N], e);
    }
  }
}

// ============================================================================
// Generic 1x1-conv GEMM: Y[b,o,n] = sum_c W[o,c]*X[b,c,n] + bias
// One wave -> 16(O) x 64(N): 4 accumulators, A fragment reused 4x.
//   STORE    : write Y
//   STATS    : accumulate per-(b,group) sum/sumsq of y (incl. bias)
//   MINMAX_O : per-(b,o) running max/min over n (global max-pool layer)
// ============================================================================
template<bool STORE, bool STATS, bool MINMAX_O>
__global__ void k_conv_gemm(const float* __restrict__ W, int ldw,
                            const float* __restrict__ X, int C, int O, int N,
                            const float* __restrict__ bias, int biasStride,
                            float* __restrict__ Y,
                            float* __restrict__ stats, int CG,
                            unsigned* __restrict__ mmax, unsigned* __restrict__ mmin) {
  int b  = blockIdx.z;
  int o0 = blockIdx.y * 16;
  int w    = threadIdx.x >> 5;
  int lane = threadIdx.x & 31;
  int h = lane >> 4, l = lane & 15;
  int n0 = (blockIdx.x * 8 + w) * 64;

  const float* Xb   = X + (size_t)b * C * N;
  const float* wrow = W + (size_t)(o0 + l) * ldw;

  v8f acc[4];
#pragma unroll
  for (int t = 0; t < 4; t++) acc[t] = (v8f){};

  for (int k0 = 0; k0 < C; k0 += 4) {           // all C are multiples of 4
    int ka = k0 + 2 * h;
    v2f a = ld2(wrow + ka);
    const float* xr = Xb + (size_t)ka * N + n0 + l;
#pragma unroll
    for (int t = 0; t < 4; t++) {
      v2f bb; bb[0] = xr[16 * t]; bb[1] = xr[(size_t)N + 16 * t];
      acc[t] = __builtin_amdgcn_wmma_f32_16x16x4_f32(false, a, false, bb,
                                                     (short)0, acc[t], false, false);
    }
  }

  float d[4][8];
#pragma unroll
  for (int t = 0; t < 4; t++)
#pragma unroll
    for (int v = 0; v < 8; v++) {
      int o = o0 + v + 8 * h;
      float bv = bias ? bias[(size_t)b * biasStride + o] : 0.f;
      d[t][v] = acc[t][v] + bv;
    }

  if (STORE) {
#pragma unroll
    for (int t = 0; t < 4; t++)
#pragma unroll
      for (int v = 0; v < 8; v++) {
        int o = o0 + v + 8 * h;
        Y[((size_t)b * O + o) * N + n0 + 16 * t + l] = d[t][v];
      }
  }
  if (STATS) {
    float s = 0.f, ss = 0.f;
#pragma unroll
    for (int t = 0; t < 4; t++)
#pragma unroll
      for (int v = 0; v < 8; v++) { s += d[t][v]; ss += d[t][v] * d[t][v]; }
    stats_reduce_atomic(s, ss, l, h, o0, CG, b, O / CG, stats);
  }
  if (MINMAX_O) {
#pragma unroll
    for (int v = 0; v < 8; v++) {
      float mx = d[0][v], mn = d[0][v];
#pragma unroll
      for (int t = 1; t < 4; t++) {
        mx = fmaxf(mx, d[t][v]);
        mn = fminf(mn, d[t][v]);
      }
#pragma unroll
      for (int m = 1; m < 16; m <<= 1) {
        mx = fmaxf(mx, __shfl_xor(mx, m, 32));
        mn = fminf(mn, __shfl_xor(mn, m, 32));
      }
      if (l == 0) {
        int o = o0 + v + 8 * h;
        atomicMax(&mmax[(size_t)b * O + o], fenc(mx));
        atomicMin(&mmin[(size_t)b * O + o], fenc(mn));
      }
    }
  }
}

// ============================================================================
// finalize kernels
// ============================================================================
// edge block: out = leaky(gn(max_k h)) using sign(gamma) max/min trick
__global__ void k_edge_finalize(const float* __restrict__ stats, float cntInv,
                                const float* __restrict__ gw, const float* __restrict__ gb,
                                int CG, int O, int N, int cbase, int CFE,
                                float* __restrict__ feats, float* __restrict__ xtout,
                                const unsigned* __restrict__ mmax,
                                const unsigned* __restrict__ mmin, int total) {
  int i = blockIdx.x * blockDim.x + threadIdx.x;
  if (i >= total) return;
  int n = i % N; int o = (i / N) % O; int b = i / (N * O);
  int g = o / CG; int G = O / CG;
  float sum = stats[2 * ((size_t)b * G + g)], ssq = stats[2 * ((size_t)b * G + g) + 1];
  float mu = sum * cntInv;
  float var = ssq * cntInv - mu * mu;
  float rs = rsqrtf(var + EPSV);
  float wv = gw[o], bv = gb[o];
  size_t mi = ((size_t)b * O + o) * N + n;
  float sel = (wv >= 0.f) ? fdec(mmax[mi]) : fdec(mmin[mi]);
  float outv = leakyf((sel - mu) * rs * wv + bv);
  feats[((size_t)b * CFE + cbase + o) * N + n] = outv;
  if (xtout) xtout[((size_t)b * N + n) * O + o] = outv;
}

// Wm layer: xmax[b,c] = relu(gn(max_n y)) via sign(gamma) trick
__global__ void k_wm_finalize(const float* __restrict__ stats, float cntInv,
                              const float* __restrict__ gw, const float* __restrict__ gb,
                              int CG, int O,
                              float* __restrict__ xmax,
                              const unsigned* __restrict__ mmax,
                              const unsigned* __restrict__ mmin, int total) {
  int i = blockIdx.x * blockDim.x + threadIdx.x;
  if (i >= total) return;
  int o = i % O; int b = i / O;
  int g = o / CG; int G = O / CG;
  float sum = stats[2 * ((size_t)b * G + g)], ssq = stats[2 * ((size_t)b * G + g) + 1];
  float mu = sum * cntInv;
  float var = ssq * cntInv - mu * mu;
  float rs = rsqrtf(var + EPSV);
  float wv = gw[o], bv = gb[o];
  float sel = (wv >= 0.f) ? fdec(mmax[i]) : fdec(mmin[i]);
  float v = (sel - mu) * rs * wv + bv;
  xmax[i] = v > 0.f ? v : 0.f;
}

// in-place gn + relu on h (B,O,N)
__global__ void k_norm_relu(float* __restrict__ hbuf,
                            const float* __restrict__ stats, float cntInv,
                            const float* __restrict__ gw, const float* __restrict__ gb,
                            int CG, int O, int N, int total) {
  int i = blockIdx.x * blockDim.x + threadIdx.x;
  if (i >= total) return;
  int o = (i / N) % O; int b = i / (N * O);
  int g = o / CG; int G = O / CG;
  float sum = stats[2 * ((size_t)b * G + g)], ssq = stats[2 * ((size_t)b * G + g) + 1];
  float mu = sum * cntInv;
  float var = ssq * cntInv - mu * mu;
  float rs = rsqrtf(var + EPSV);
  float v = (hbuf[i] - mu) * rs * gw[o] + gb[o];
  hbuf[i] = v > 0.f ? v : 0.f;
}

// v1[b,o] = bs1[o] + Ws1[o, 0:1024] . xmax[b,:]
__global__ void k_v1(const float* __restrict__ Ws1, const float* __restrict__ bs1,
                     const float* __restrict__ xmax, float* __restrict__ v1, int total) {
  int i = blockIdx.x * blockDim.x + threadIdx.x;
  if (i >= total) return;
  int o = i % 512; int b = i / 512;
  float acc = bs1[o];
  const float* wr = Ws1 + (size_t)o * 1280;
  const float* xm = xmax + (size_t)b * 1024;
  for (int c = 0; c < 1024; c++) acc += wr[c] * xm[c];
  v1[i] = acc;
}

// logits + log_softmax over 50 classes, one thread per (b,n)
__global__ void k_logits(const float* __restrict__ Ws4, const float* __restrict__ bs4,
                         const float* __restrict__ h3, float* __restrict__ out,
                         int B, int N) {
  int i = blockIdx.x * blockDim.x + threadIdx.x;
  if (i >= B * N) return;
  int n = i % N; int b = i / N;
  const float* X = h3 + (size_t)b * 128 * N;
  float acc[50];
#pragma unroll
  for (int o = 0; o < 50; o++) acc[o] = bs4[o];
  for (int c = 0; c < 128; c++) {
    float xv = X[(size_t)c * N + n];
#pragma unroll
    for (int o = 0; o < 50; o++) acc[o] += Ws4[o * 128 + c] * xv;
  }
  float m = acc[0];
#pragma unroll
  for (int o = 1; o < 50; o++) m = fmaxf(m, acc[o]);
  float se = 0.f;
#pragma unroll
  for (int o = 0; o < 50; o++) se += __expf(acc[o] - m);
  float lse = m + __logf(se);
#pragma unroll
  for (int o = 0; o < 50; o++) out[((size_t)b * 50 + o) * N + n] = acc[o] - lse;
}

// ============================================================================
// launch
// ============================================================================
extern "C" void kernel_launch(void* const* d_in, const int* in_sizes, int n_in,
                              void* d_out, int out_size, void* d_ws, size_t ws_size,
                              hipStream_t stream) {
  (void)in_sizes; (void)n_in; (void)out_size; (void)ws_size;
  const int B = 8, N = 2048;

  const float* x    = (const float*)d_in[0];
  const float* W1   = (const float*)d_in[1];
  const float* W2   = (const float*)d_in[2];
  const float* W3   = (const float*)d_in[3];
  const float* g1w  = (const float*)d_in[4];  const float* g1b = (const float*)d_in[5];
  const float* g2w  = (const float*)d_in[6];  const float* g2b = (const float*)d_in[7];
  const float* g3w  = (const float*)d_in[8];  const float* g3b = (const float*)d_in[9];
  const float* gfw  = (const float*)d_in[10]; const float* gfb = (const float*)d_in[11];
  const float* gs1w = (const float*)d_in[12]; const float* gs1b = (const float*)d_in[13];
  const float* gs2w = (const float*)d_in[14]; const float* gs2b = (const float*)d_in[15];
  const float* gs3w = (const float*)d_in[16]; const float* gs3b = (const float*)d_in[17];
  const float* Wm   = (const float*)d_in[18]; const float* bm  = (const float*)d_in[19];
  const float* Ws1  = (const float*)d_in[20]; const float* bs1 = (const float*)d_in[21];
  const float* Ws2  = (const float*)d_in[22]; const float* bs2 = (const float*)d_in[23];
  const float* Ws3  = (const float*)d_in[24]; const float* bs3 = (const float*)d_in[25];
  const float* Ws4  = (const float*)d_in[26]; const float* bs4 = (const float*)d_in[27];

  float* ws = (float*)d_ws;
  // workspace layout (float element offsets)
  const size_t XT0   = 0;                      //  B*N*3
  const size_t XT1   = XT0 + 49152;            //  B*N*64
  const size_t XT2   = XT1 + 1048576;          //  B*N*64
  const size_t FEATS = XT2 + 1048576;          //  B*256*N
  const size_t H1    = FEATS + 4194304;        //  B*512*N
  const size_t H2    = H1 + 8388608;           //  B*256*N
  const size_t H3    = H2 + 4194304;           //  B*128*N
  const size_t XMAX  = H3 + 2097152;           //  B*1024
  const size_t V1    = XMAX + 8192;            //  B*512
  const size_t STATS = V1 + 4096;              //  2048 floats
  const size_t IDXO  = STATS + 2048;           //  B*N*KNN ints
  const size_t MM    = IDXO + 327680;          //  2 * 2097152 uints
  const int MMHALF = 2097152;

  int*      idx = (int*)(ws + IDXO);
  unsigned* mmx = (unsigned*)(ws + MM);
  unsigned* mmn = mmx + MMHALF;

  // stats slices (float offsets within STATS)
  float* SE1 = ws + STATS + 0;    // B*8  groups
  float* SE2 = ws + STATS + 128;
  float* SE3 = ws + STATS + 256;  // B*8
  float* SWM = ws + STATS + 384;  // B*32
  float* SG1 = ws + STATS + 896;  // B*16
  float* SG2 = ws + STATS + 1152;
  float* SG3 = ws + STATS + 1408; // B*8

  const int T = 256;
  const int colBlocks = (N * KNN) / (64 * 8);   // 80 blocks.x for edge GEMMs
  const int nBlocks   = N / (64 * 8);           // 4  blocks.x for conv GEMMs

  k_zero_f<<<(2048 + T - 1) / T, T, 0, stream>>>(ws + STATS, 2048);
  k_transpose3<<<(B * 3 * N + T - 1) / T, T, 0, stream>>>(x, ws + XT0, B, N);

  // ---- edge block 1 (C=3, O=64, G=8) ----
  k_knn<3><<<B * N / 8, T, 0, stream>>>(ws + XT0, idx, B, N);
  k_init_mm<<<(B * 64 * N + T - 1) / T, T, 0, stream>>>(mmx, MMHALF, B * 64 * N);
  k_edge_gemm<3><<<dim3(colBlocks, 4, B), T, 0, stream>>>(W1, ws + XT0, idx,
                                                          64, N, SE1, 8, mmx, mmn);
  k_edge_finalize<<<(B * 64 * N + T - 1) / T, T, 0, stream>>>(
      SE1, 1.f / (8.f * N * KNN), g1w, g1b, 8, 64, N, 0, 256,
      ws + FEATS, ws + XT1, mmx, mmn, B * 64 * N);

  // ---- edge block 2 (C=64, O=64, G=8) ----
  k_knn<64><<<B * N / 8, T, 0, stream>>>(ws + XT1, idx, B, N);
  k_init_mm<<<(B * 64 * N + T - 1) / T, T, 0, stream>>>(mmx, MMHALF, B * 64 * N);
  k_edge_gemm<64><<<dim3(colBlocks, 4, B), T, 0, stream>>>(W2, ws + XT1, idx,
                                                           64, N, SE2, 8, mmx, mmn);
  k_edge_finalize<<<(B * 64 * N + T - 1) / T, T, 0, stream>>>(
      SE2, 1.f / (8.f * N * KNN), g2w, g2b, 8, 64, N, 64, 256,
      ws + FEATS, ws + XT2, mmx, mmn, B * 64 * N);

  // ---- edge block 3 (C=64, O=128, G=8 -> CG=16) ----
  k_knn<64><<<B * N / 8, T, 0, stream>>>(ws + XT2, idx, B, N);
  k_init_mm<<<(B * 128 * N + T - 1) / T, T, 0, stream>>>(mmx, MMHALF, B * 128 * N);
  k_edge_gemm<64><<<dim3(colBlocks, 8, B), T, 0, stream>>>(W3, ws + XT2, idx,
                                                           128, N, SE3, 16, mmx, mmn);
  k_edge_finalize<<<(B * 128 * N + T - 1) / T, T, 0, stream>>>(
      SE3, 1.f / (16.f * N * KNN), g3w, g3b, 16, 128, N, 128, 256,
      ws + FEATS, nullptr, mmx, mmn, B * 128 * N);

  // ---- Wm (1024x256) + gn(32) + relu + max over n  (stats+minmax pass only) ----
  k_init_mm<<<(B * 1024 + T - 1) / T, T, 0, stream>>>(mmx, MMHALF, B * 1024);
  k_conv_gemm<false, true, true><<<dim3(nBlocks, 64, B), T, 0, stream>>>(
      Wm, 256, ws + FEATS, 256, 1024, N, bm, 0, nullptr, SWM, 32, mmx, mmn);
  k_wm_finalize<<<(B * 1024 + T - 1) / T, T, 0, stream>>>(
      SWM, 1.f / (32.f * N), gfw, gfb, 32, 1024, ws + XMAX, mmx, mmn, B * 1024);

  // ---- Ws1 = Ws1[:, :1024]@xmax (per batch)  +  Ws1[:, 1024:]@feats ----
  k_v1<<<(B * 512 + T - 1) / T, T, 0, stream>>>(Ws1, bs1, ws + XMAX, ws + V1, B * 512);
  k_conv_gemm<true, true, false><<<dim3(nBlocks, 32, B), T, 0, stream>>>(
      Ws1 + 1024, 1280, ws + FEATS, 256, 512, N, ws + V1, 512, ws + H1,
      SG1, 32, nullptr, nullptr);
  k_norm_relu<<<(B * 512 * N + T - 1) / T, T, 0, stream>>>(
      ws + H1, SG1, 1.f / (32.f * N), gs1w, gs1b, 32, 512, N, B * 512 * N);

  // ---- Ws2 (256x512) + gn(16) + relu ----
  k_conv_gemm<true, true, false><<<dim3(nBlocks, 16, B), T, 0, stream>>>(
      Ws2, 512, ws + H1, 512, 256, N, bs2, 0, ws + H2, SG2, 16, nullptr, nullptr);
  k_norm_relu<<<(B * 256 * N + T - 1) / T, T, 0, stream>>>(
      ws + H2, SG2, 1.f / (16.f * N), gs2w, gs2b, 16, 256, N, B * 256 * N);

  // ---- Ws3 (128x256) + gn(8) + relu ----
  k_conv_gemm<true, true, false><<<dim3(nBlocks, 8, B), T, 0, stream>>>(
      Ws3, 256, ws + H2, 256, 128, N, bs3, 0, ws + H3, SG3, 16, nullptr, nullptr);
  k_norm_relu<<<(B * 128 * N + T - 1) / T, T, 0, stream>>>(
      ws + H3, SG3, 1.f / (16.f * N), gs3w, gs3b, 16, 128, N, B * 128 * N);

  // ---- logits + log_softmax ----
  k_logits<<<(B * N + T - 1) / T, T, 0, stream>>>(Ws4, bs4, ws + H3,
                                                  (float*)d_out, B, N);
}